// Bottleneck_25778393710891
// MI455X (gfx1250) — compile-verified
//
#include <hip/hip_runtime.h>
#include <math.h>

#define N_NODES 50000
#define N_EDGES 800000
#define D_IN    256
#define D_OUT   64
#define BN_EPS  1e-5f

typedef float v2f __attribute__((ext_vector_type(2)));
typedef float v8f __attribute__((ext_vector_type(8)));

static __device__ __forceinline__ v8f wmma_f32_k4(v2f a, v2f b, v8f c) {
  return __builtin_amdgcn_wmma_f32_16x16x4_f32(
      /*neg_a=*/false, a, /*neg_b=*/false, b,
      /*c_mod=*/(short)0, c, /*reuse_a=*/false, /*reuse_b=*/false);
}

// ---------------- degree / dinv / stat-zero ----------------
__global__ void k_init(float* __restrict__ deg, float* __restrict__ stats) {
  int i = blockIdx.x * 256 + threadIdx.x;
  if (i < N_NODES) deg[i] = 1.0f;     // self-loop
  if (i < 768)     stats[i] = 0.0f;   // all BN stat accumulators
}

__global__ void k_edge_deg(const int* __restrict__ dst, float* __restrict__ deg) {
  int i = blockIdx.x * 256 + threadIdx.x;
  if (i < N_EDGES) atomicAdd(&deg[dst[i]], 1.0f);
}

__global__ void k_dinv(float* __restrict__ deg) {
  int i = blockIdx.x * 256 + threadIdx.x;
  if (i < N_NODES) deg[i] = rsqrtf(deg[i]);
}

// ---------------- GEMM: H = X @ W ; AGG = H*dinv^2 + bias ----------------
// One wave computes a 16-row x 64-col tile (4 WMMA accumulators).
// W is staged to LDS transposed in 64-K chunks: Wt[col][k] so B fragments are
// contiguous 8-byte ds loads. f32 WMMA 16x16x4, K loop of KD/4 steps.
template <int KD, int ND>
__global__ __launch_bounds__(256) void k_gemm_wmma(
    const float* __restrict__ X, const float* __restrict__ W,
    const float* __restrict__ bias, const float* __restrict__ dinv,
    float* __restrict__ H, float* __restrict__ AGG)
{
  constexpr int KC = 64;                  // K staging chunk
  __shared__ float Wt[64 * KC];           // 16 KB: Wt[col][k] (col-major slice)

  const int tid  = threadIdx.x;
  const int wave = tid >> 5;              // wave32
  const int lane = tid & 31;
  const int half = lane >> 4;             // 0: K even-pair low half, 1: high half
  const int l16  = lane & 15;

  const int nTiles = N_NODES / 16;        // 3125 (exact)
  const int tile   = blockIdx.x * 8 + wave;
  const bool valid = (tile < nTiles);     // wave-uniform -> EXEC stays all-ones
  const int r0 = tile * 16;
  const int c0 = blockIdx.y * 64;

  v8f acc[4] = {};

  const int arow = r0 + l16;              // A-matrix row for this lane

  for (int kb = 0; kb < KD; kb += KC) {
    __syncthreads();
    // stage W[kb..kb+KC) x [c0..c0+64) transposed into LDS
    for (int i = tid; i < 64 * KC; i += 256) {
      int kk = i >> 6;
      int c  = i & 63;
      Wt[c * KC + kk] = W[(size_t)(kb + kk) * ND + c0 + c];
    }
    __syncthreads();

    if (valid) {
      const float* xrow = X + (size_t)arow * KD + kb + 2 * half;
      #pragma unroll 4
      for (int k0 = 0; k0 < KC; k0 += 4) {
        // A 16x4 f32 fragment: lane holds {X[row][k0+2h], X[row][k0+2h+1]}
        v2f a = *(const v2f*)(xrow + k0);
        __builtin_prefetch(xrow + k0 + 16, 0, 1);
        #pragma unroll
        for (int c = 0; c < 4; ++c) {
          // B 4x16 fragment: lane holds {W[k0+2h][col], W[k0+2h+1][col]}
          v2f b = *(const v2f*)&Wt[(c * 16 + l16) * KC + k0 + 2 * half];
          acc[c] = wmma_f32_k4(a, b, acc[c]);
        }
      }
    }
  }

  if (valid) {
    // D 16x16 f32 layout: VGPR j -> M=j (lanes 0-15), M=j+8 (lanes 16-31), N=l16
    #pragma unroll
    for (int c = 0; c < 4; ++c) {
      #pragma unroll
      for (int j = 0; j < 8; ++j) {
        int row = r0 + j + 8 * half;
        int col = c0 + c * 16 + l16;
        float v  = acc[c][j];
        float di = dinv[row];
        H  [(size_t)row * ND + col] = v;
        AGG[(size_t)row * ND + col] = v * di * di + bias[col];  // self-loop + bias
      }
    }
  }
}

// ---------------- edge scatter: AGG[dst] += H[src] * dinv[src]*dinv[dst] ----
template <int C>
__global__ void k_scatter(const int* __restrict__ src, const int* __restrict__ dst,
                          const float* __restrict__ dinv, const float* __restrict__ H,
                          float* __restrict__ AGG)
{
  unsigned idx = blockIdx.x * 256u + threadIdx.x;
  unsigned e = idx / C;
  if (e >= N_EDGES) return;
  unsigned c = idx & (C - 1);
  int s = src[e], d = dst[e];
  float nrm = dinv[s] * dinv[d];
  atomicAdd(&AGG[(size_t)d * C + c], H[(size_t)s * C + c] * nrm);
}

// ---------------- BN stats: per-column sum / sumsq ----------------
template <int C>
__global__ void k_bn_stats(const float* __restrict__ A,
                           float* __restrict__ sum, float* __restrict__ sq)
{
  const int ROWS = 128;
  int col = threadIdx.x;                  // blockDim.x == C
  int r0  = blockIdx.x * ROWS;
  float s = 0.f, q = 0.f;
  for (int i = 0; i < ROWS; ++i) {
    int r = r0 + i;
    if (r < N_NODES) {
      float v = A[(size_t)r * C + col];
      s += v; q += v * v;
    }
  }
  atomicAdd(&sum[col], s);
  atomicAdd(&sq[col], q);
}

// ---------------- BN apply (+optional residual, ReLU), in place -------------
template <int C, bool RELU, bool RES>
__global__ void k_bn_apply(float* __restrict__ A,
                           const float* __restrict__ sum, const float* __restrict__ sq,
                           const float* __restrict__ g, const float* __restrict__ be,
                           const float* __restrict__ res)
{
  unsigned idx = blockIdx.x * 256u + threadIdx.x;
  if (idx >= (unsigned)N_NODES * C) return;
  unsigned col = idx & (C - 1);
  const float invn = 1.0f / (float)N_NODES;
  float mean = sum[col] * invn;
  float var  = sq[col] * invn - mean * mean;
  float sc   = g[col] * rsqrtf(var + BN_EPS);
  float v = (A[idx] - mean) * sc + be[col];
  if (RES)  v += res[idx];
  if (RELU) v = fmaxf(v, 0.0f);
  A[idx] = v;
}

// ---------------- host ----------------
static inline unsigned cdiv(unsigned a, unsigned b) { return (a + b - 1) / b; }

extern "C" void kernel_launch(void* const* d_in, const int* in_sizes, int n_in,
                              void* d_out, int out_size, void* d_ws, size_t ws_size,
                              hipStream_t stream) {
  const float* x   = (const float*)d_in[0];
  const int*   ei  = (const int*)d_in[1];
  const int*   src = ei;
  const int*   dst = ei + N_EDGES;
  const float* W1 = (const float*)d_in[3],  *b1 = (const float*)d_in[4];
  const float* g1 = (const float*)d_in[5],  *be1 = (const float*)d_in[6];
  const float* W2 = (const float*)d_in[7],  *b2 = (const float*)d_in[8];
  const float* g2 = (const float*)d_in[9],  *be2 = (const float*)d_in[10];
  const float* W3 = (const float*)d_in[11], *b3 = (const float*)d_in[12];
  const float* g3 = (const float*)d_in[13], *be3 = (const float*)d_in[14];
  float* out = (float*)d_out;

  // workspace layout (floats)
  float* ws    = (float*)d_ws;
  float* dinv  = ws;                                 // N
  float* hbig  = dinv + N_NODES;                     // N*256 (h3)
  float* hsml  = hbig + (size_t)N_NODES * 256;       // N*64  (h1, then h2)
  float* agg1  = hsml + (size_t)N_NODES * 64;        // N*64
  float* agg2  = agg1 + (size_t)N_NODES * 64;        // N*64
  float* stats = agg2 + (size_t)N_NODES * 64;        // 768
  float* sum1 = stats,       *sq1 = stats + 64;
  float* sum2 = stats + 128, *sq2 = stats + 192;
  float* sum3 = stats + 256, *sq3 = stats + 512;

  const unsigned nbN = cdiv(N_NODES, 256);
  const unsigned nbE = cdiv(N_EDGES, 256);
  const unsigned gemmBlks = cdiv(N_NODES / 16, 8);   // 391
  const unsigned statBlks = cdiv(N_NODES, 128);      // 391

  // degree with self-loops -> dinv; zero BN stats
  k_init<<<nbN, 256, 0, stream>>>(dinv, stats);
  k_edge_deg<<<nbE, 256, 0, stream>>>(dst, dinv);
  k_dinv<<<nbN, 256, 0, stream>>>(dinv);

  // ---- layer 1: x[ N,256 ] @ W1 -> 64 ----
  k_gemm_wmma<D_IN, 64><<<dim3(gemmBlks, 1), 256, 0, stream>>>(x, W1, b1, dinv, hsml, agg1);
  k_scatter<64><<<cdiv((unsigned)N_EDGES * 64, 256), 256, 0, stream>>>(src, dst, dinv, hsml, agg1);
  k_bn_stats<64><<<statBlks, 64, 0, stream>>>(agg1, sum1, sq1);
  k_bn_apply<64, true, false><<<cdiv((unsigned)N_NODES * 64, 256), 256, 0, stream>>>(
      agg1, sum1, sq1, g1, be1, nullptr);

  // ---- layer 2: y1[ N,64 ] @ W2 -> 64 ----
  k_gemm_wmma<64, 64><<<dim3(gemmBlks, 1), 256, 0, stream>>>(agg1, W2, b2, dinv, hsml, agg2);
  k_scatter<64><<<cdiv((unsigned)N_EDGES * 64, 256), 256, 0, stream>>>(src, dst, dinv, hsml, agg2);
  k_bn_stats<64><<<statBlks, 64, 0, stream>>>(agg2, sum2, sq2);
  k_bn_apply<64, true, false><<<cdiv((unsigned)N_NODES * 64, 256), 256, 0, stream>>>(
      agg2, sum2, sq2, g2, be2, nullptr);

  // ---- layer 3: y2[ N,64 ] @ W3 -> 256 ; agg3 lives in d_out ----
  k_gemm_wmma<64, 256><<<dim3(gemmBlks, 4), 256, 0, stream>>>(agg2, W3, b3, dinv, hbig, out);
  k_scatter<256><<<cdiv((unsigned)N_EDGES * 256, 256), 256, 0, stream>>>(src, dst, dinv, hbig, out);
  k_bn_stats<256><<<statBlks, 256, 0, stream>>>(out, sum3, sq3);
  k_bn_apply<256, true, true><<<cdiv((unsigned)N_NODES * 256, 256), 256, 0, stream>>>(
      out, sum3, sq3, g3, be3, x);
}